// DOGEPredictor_21784074125681
// MI455X (gfx1250) — compile-verified
//
#include <hip/hip_runtime.h>
#include <math.h>

typedef float v2f __attribute__((ext_vector_type(2)));
typedef float v8f __attribute__((ext_vector_type(8)));

// D = A(16x4 f32) * B(4x16 f32) + C(16x16 f32), full fp32 precision on the matrix pipe.
static __device__ __forceinline__ v8f wmma4(v2f a, v2f b, v8f c) {
  return __builtin_amdgcn_wmma_f32_16x16x4_f32(false, a, false, b, (short)0, c, false, false);
}

static __device__ __forceinline__ void atomicMaxF(float* a, float v) {
  if (v >= 0.0f) atomicMax((int*)a, __float_as_int(v));
  else           atomicMin((unsigned int*)a, __float_as_uint(v));
}

// Pair-interleaved weight layout: pair p holds rows 2p,2p+1 adjacent per column:
//   sW[p*128 + col*2 + (r&1)]  ->  B-fragment = one aligned ds_load_b64.

// ---------------------------------------------------------------- utilities
__global__ void fill_kernel(float* __restrict__ p, float v, long n) {
  long i = (long)blockIdx.x * blockDim.x + threadIdx.x;
  if (i < n) p[i] = v;
}

// interleave a [64,64] weight into pair layout in ws (for global B-frag b64 loads)
__global__ void interleave_w_kernel(const float* __restrict__ W, float* __restrict__ Wp) {
  int idx = blockIdx.x * blockDim.x + threadIdx.x;
  if (idx >= 64 * 64) return;
  int r = idx >> 6, c = idx & 63;
  Wp[(r >> 1) * 128 + c * 2 + (r & 1)] = W[idx];
}

// y[n,64] = x[n,F] @ W[F,64] + b   (tiny K, VALU)
__global__ void node_linear_kernel(const float* __restrict__ x, int F,
                                   const float* __restrict__ W, const float* __restrict__ b,
                                   float* __restrict__ y, long total) {
  long idx = (long)blockIdx.x * blockDim.x + threadIdx.x;
  if (idx >= total) return;
  int n = (int)(idx >> 6), d = (int)(idx & 63);
  float acc = b[d];
  for (int j = 0; j < F; ++j) acc = fmaf(x[(size_t)n * F + j], W[j * 64 + d], acc);
  y[idx] = acc;
}

// out = relu(numer/max(denom,1e-16)/max(deg,1) + x@Wsk + bsk)
__global__ void finalize_kernel(const float* __restrict__ numer, const float* __restrict__ denom,
                                const float* __restrict__ deg, const float* __restrict__ x, int F,
                                const float* __restrict__ Wsk, const float* __restrict__ bsk,
                                float* __restrict__ out, long total) {
  long idx = (long)blockIdx.x * blockDim.x + threadIdx.x;
  if (idx >= total) return;
  int n = (int)(idx >> 6), d = (int)(idx & 63);
  float dn = fmaxf(denom[n], 1e-16f);
  float dg = fmaxf(deg[n], 1.0f);
  float agg = numer[idx] / dn / dg;
  float s = bsk[d];
  for (int j = 0; j < F; ++j) s = fmaf(x[(size_t)n * F + j], Wsk[j * 64 + d], s);
  out[idx] = fmaxf(agg + s, 0.0f);
}

// ------------------------------------------- con_conv (dst = constraints) passes
__global__ __launch_bounds__(256) void con_pass1_kernel(
    const float* __restrict__ var_lp,
    const float* __restrict__ lo, const float* __restrict__ hi,
    const float* __restrict__ dmm, const float* __restrict__ elp,
    const int* __restrict__ vi, const int* __restrict__ ci,
    const float* __restrict__ qc,
    const float* __restrict__ Wk, const float* __restrict__ bk,
    const float* __restrict__ We, const float* __restrict__ be,
    float* __restrict__ logits, float* __restrict__ mA, float* __restrict__ degA, int E) {
  __shared__ float sW[512];  // Wk[2*64] | We[5*64] | (bk+be)[64]
  int tid = threadIdx.x;
  for (int i = tid; i < 448; i += 256) sW[i] = (i < 128) ? Wk[i] : We[i - 128];
  for (int i = tid; i < 64; i += 256) sW[448 + i] = bk[i] + be[i];
  __syncthreads();
  int e = blockIdx.x * 256 + tid;
  if (e >= E) return;
  int v = vi[e], c = ci[e];
  float a0 = var_lp[(size_t)v * 2 + 0], a1 = var_lp[(size_t)v * 2 + 1];
  float f0 = lo[e], f1 = hi[e], f2 = dmm[e], f3 = elp[(size_t)e * 2], f4 = elp[(size_t)e * 2 + 1];
  const float4* q4 = (const float4*)(qc + (size_t)c * 64);
  float acc = 0.0f;
  auto keF = [&](int d) {
    return sW[448 + d] + a0 * sW[d] + a1 * sW[64 + d] + f0 * sW[128 + d] + f1 * sW[192 + d] +
           f2 * sW[256 + d] + f3 * sW[320 + d] + f4 * sW[384 + d];
  };
#pragma unroll
  for (int dd = 0; dd < 16; ++dd) {
    float4 qv = q4[dd];
    int d = dd * 4;
    acc = fmaf(qv.x, keF(d + 0), acc);
    acc = fmaf(qv.y, keF(d + 1), acc);
    acc = fmaf(qv.z, keF(d + 2), acc);
    acc = fmaf(qv.w, keF(d + 3), acc);
  }
  float lg = acc * 0.125f;
  logits[e] = lg;
  atomicMaxF(&mA[c], lg);
  atomicAdd(&degA[c], 1.0f);
}

__global__ __launch_bounds__(256) void con_pass2_kernel(
    const float* __restrict__ var_lp,
    const float* __restrict__ lo, const float* __restrict__ hi,
    const float* __restrict__ dmm, const float* __restrict__ elp,
    const int* __restrict__ vi, const int* __restrict__ ci,
    const float* __restrict__ Wv, const float* __restrict__ bv,
    const float* __restrict__ We, const float* __restrict__ be,
    const float* __restrict__ logits, const float* __restrict__ mA,
    float* __restrict__ numerA, float* __restrict__ denomA, int E) {
  __shared__ float sW[512];
  int tid = threadIdx.x;
  for (int i = tid; i < 448; i += 256) sW[i] = (i < 128) ? Wv[i] : We[i - 128];
  for (int i = tid; i < 64; i += 256) sW[448 + i] = bv[i] + be[i];
  __syncthreads();
  int e = blockIdx.x * 256 + tid;
  if (e >= E) return;
  int v = vi[e], c = ci[e];
  float a0 = var_lp[(size_t)v * 2 + 0], a1 = var_lp[(size_t)v * 2 + 1];
  float f0 = lo[e], f1 = hi[e], f2 = dmm[e], f3 = elp[(size_t)e * 2], f4 = elp[(size_t)e * 2 + 1];
  float w = __expf(logits[e] - mA[c]);
  float* nrow = numerA + (size_t)c * 64;
#pragma unroll
  for (int d = 0; d < 64; ++d) {
    float ve = sW[448 + d] + a0 * sW[d] + a1 * sW[64 + d] + f0 * sW[128 + d] + f1 * sW[192 + d] +
               f2 * sW[256 + d] + f3 * sW[320 + d] + f4 * sW[384 + d];
    atomicAdd(&nrow[d], w * ve);
  }
  atomicAdd(&denomA[c], w);
}

// ------------------------------------------- var_conv (dst = variables) passes
__global__ __launch_bounds__(256) void var_pass1_kernel(
    const float* __restrict__ var_lp,
    const float* __restrict__ lo, const float* __restrict__ hi,
    const float* __restrict__ dmm, const float* __restrict__ elp,
    const int* __restrict__ vi, const int* __restrict__ ci,
    const float* __restrict__ k2,
    const float* __restrict__ Wq, const float* __restrict__ bq,
    const float* __restrict__ We, const float* __restrict__ be,
    float* __restrict__ logits, float* __restrict__ mB, float* __restrict__ degB, int E) {
  __shared__ float sW[576];  // Wq[2*64] | We[5*64] | bq[64] | be[64]
  int tid = threadIdx.x;
  for (int i = tid; i < 448; i += 256) sW[i] = (i < 128) ? Wq[i] : We[i - 128];
  for (int i = tid; i < 64; i += 256) { sW[448 + i] = bq[i]; sW[512 + i] = be[i]; }
  __syncthreads();
  int e = blockIdx.x * 256 + tid;
  if (e >= E) return;
  int v = vi[e], c = ci[e];
  float a0 = var_lp[(size_t)v * 2 + 0], a1 = var_lp[(size_t)v * 2 + 1];
  float f0 = lo[e], f1 = hi[e], f2 = dmm[e], f3 = elp[(size_t)e * 2], f4 = elp[(size_t)e * 2 + 1];
  const float4* k4 = (const float4*)(k2 + (size_t)c * 64);
  float acc = 0.0f;
  auto eF = [&](int d) {
    return sW[512 + d] + f0 * sW[128 + d] + f1 * sW[192 + d] + f2 * sW[256 + d] +
           f3 * sW[320 + d] + f4 * sW[384 + d];
  };
  auto qF = [&](int d) { return sW[448 + d] + a0 * sW[d] + a1 * sW[64 + d]; };
#pragma unroll
  for (int dd = 0; dd < 16; ++dd) {
    float4 kv = k4[dd];
    int d = dd * 4;
    acc = fmaf(qF(d + 0), kv.x + eF(d + 0), acc);
    acc = fmaf(qF(d + 1), kv.y + eF(d + 1), acc);
    acc = fmaf(qF(d + 2), kv.z + eF(d + 2), acc);
    acc = fmaf(qF(d + 3), kv.w + eF(d + 3), acc);
  }
  float lg = acc * 0.125f;
  logits[e] = lg;
  atomicMaxF(&mB[v], lg);
  atomicAdd(&degB[v], 1.0f);
}

__global__ __launch_bounds__(256) void var_pass2_kernel(
    const float* __restrict__ lo, const float* __restrict__ hi,
    const float* __restrict__ dmm, const float* __restrict__ elp,
    const int* __restrict__ vi, const int* __restrict__ ci,
    const float* __restrict__ v2,
    const float* __restrict__ We, const float* __restrict__ be,
    const float* __restrict__ logits, const float* __restrict__ mB,
    float* __restrict__ numerB, float* __restrict__ denomB, int E) {
  __shared__ float sW[384];  // We[5*64] | be[64]
  int tid = threadIdx.x;
  for (int i = tid; i < 320; i += 256) sW[i] = We[i];
  for (int i = tid; i < 64; i += 256) sW[320 + i] = be[i];
  __syncthreads();
  int e = blockIdx.x * 256 + tid;
  if (e >= E) return;
  int v = vi[e], c = ci[e];
  float f0 = lo[e], f1 = hi[e], f2 = dmm[e], f3 = elp[(size_t)e * 2], f4 = elp[(size_t)e * 2 + 1];
  float w = __expf(logits[e] - mB[v]);
  const float* vrow = v2 + (size_t)c * 64;
  float* nrow = numerB + (size_t)v * 64;
#pragma unroll
  for (int d = 0; d < 64; ++d) {
    float ve = vrow[d] + sW[320 + d] + f0 * sW[d] + f1 * sW[64 + d] + f2 * sW[128 + d] +
               f3 * sW[192 + d] + f4 * sW[256 + d];
    atomicAdd(&nrow[d], w * ve);
  }
  atomicAdd(&denomB[v], w);
}

// ---------------------------------------------------------------- WMMA GEMM kernels
#define KPMAX 72
#define NTILES 4

// k2/v2 = [learned(64)|lp(F1)] @ {Wk,Wv} + {bk,bv}  (no relu); 4 waves x 4 tiles x 16 rows
__global__ __launch_bounds__(128) void dual_linear_wmma_kernel(
    const float* __restrict__ learned, const float* __restrict__ lp, int F1, int N,
    const float* __restrict__ Wk, const float* __restrict__ bk,
    const float* __restrict__ Wv, const float* __restrict__ bv,
    float* __restrict__ outK, float* __restrict__ outV) {
  const int K = 64 + F1;
  const int KP = (K + 3) & ~3;
  __shared__ alignas(16) float sWk[KPMAX * 64];
  __shared__ alignas(16) float sWv[KPMAX * 64];
  __shared__ float sBk[64], sBv[64];
  __shared__ alignas(16) float sA[4][16 * KPMAX];
  int tid = threadIdx.x;
  for (int i = tid; i < KP * 64; i += 128) {
    int r = i >> 6, c = i & 63;
    int pi = (r >> 1) * 128 + c * 2 + (r & 1);  // pair-interleaved
    sWk[pi] = (r < K) ? Wk[r * 64 + c] : 0.0f;
    sWv[pi] = (r < K) ? Wv[r * 64 + c] : 0.0f;
  }
  if (tid < 64) { sBk[tid] = bk[tid]; sBv[tid] = bv[tid]; }
  int wave = tid >> 5, lane = tid & 31;
  int h = lane >> 4, m = lane & 15;
  float* A = sA[wave];
  __syncthreads();
  for (int t = 0; t < NTILES; ++t) {
    int base = ((blockIdx.x * 4 + wave) * NTILES + t) * 16;
    for (int r = 0; r < 16; ++r) {
      int g = base + r;
      for (int c = lane; c < KP; c += 32) {
        float val = 0.0f;
        if (g < N) {
          if (c < 64) val = learned[(size_t)g * 64 + c];
          else if (c < K) val = lp[(size_t)g * F1 + (c - 64)];
        }
        A[r * KP + c] = val;
      }
    }
    v8f ak[4] = {}, av[4] = {};
    for (int k0 = 0; k0 < KP; k0 += 4) {
      v2f a = *(const v2f*)(A + m * KP + k0 + 2 * h);
      int pr = ((k0 >> 1) + h) * 128 + m * 2;
#pragma unroll
      for (int n = 0; n < 4; ++n) {
        v2f b = *(const v2f*)(sWk + pr + n * 32);
        ak[n] = wmma4(a, b, ak[n]);
        b = *(const v2f*)(sWv + pr + n * 32);
        av[n] = wmma4(a, b, av[n]);
      }
    }
#pragma unroll
    for (int n = 0; n < 4; ++n) {
      int col = n * 16 + m;
      float bbk = sBk[col], bbv = sBv[col];
#pragma unroll
      for (int i = 0; i < 8; ++i) {
        int g = base + i + 8 * h;
        if (g < N) {
          outK[(size_t)g * 64 + col] = ak[n][i] + bbk;
          outV[(size_t)g * 64 + col] = av[n][i] + bbv;
        }
      }
    }
  }
}

// out = relu(relu([learned|lp]@W0+b0)@W1+b1); 4 waves x 4 tiles x 16 rows
__global__ __launch_bounds__(128) void mlp2_wmma_kernel(
    const float* __restrict__ learned, const float* __restrict__ lp, int F1, int N,
    const float* __restrict__ W0, const float* __restrict__ b0,
    const float* __restrict__ W1, const float* __restrict__ b1, float* __restrict__ out) {
  const int K = 64 + F1;
  const int KP = (K + 3) & ~3;
  __shared__ alignas(16) float sW0[KPMAX * 64];
  __shared__ alignas(16) float sW1[64 * 64];
  __shared__ float sB0[64], sB1[64];
  __shared__ alignas(16) float sA[4][16 * KPMAX];
  int tid = threadIdx.x;
  for (int i = tid; i < KP * 64; i += 128) {
    int r = i >> 6, c = i & 63;
    sW0[(r >> 1) * 128 + c * 2 + (r & 1)] = (r < K) ? W0[r * 64 + c] : 0.0f;
  }
  for (int i = tid; i < 64 * 64; i += 128) {
    int r = i >> 6, c = i & 63;
    sW1[(r >> 1) * 128 + c * 2 + (r & 1)] = W1[i];
  }
  if (tid < 64) { sB0[tid] = b0[tid]; sB1[tid] = b1[tid]; }
  int wave = tid >> 5, lane = tid & 31;
  int h = lane >> 4, m = lane & 15;
  float* A = sA[wave];
  __syncthreads();
  for (int t = 0; t < NTILES; ++t) {
    int base = ((blockIdx.x * 4 + wave) * NTILES + t) * 16;
    for (int r = 0; r < 16; ++r) {
      int g = base + r;
      for (int c = lane; c < KP; c += 32) {
        float val = 0.0f;
        if (g < N) {
          if (c < 64) val = learned[(size_t)g * 64 + c];
          else if (c < K) val = lp[(size_t)g * F1 + (c - 64)];
        }
        A[r * KP + c] = val;
      }
    }
    v8f acc[4] = {};
    for (int k0 = 0; k0 < KP; k0 += 4) {
      v2f a = *(const v2f*)(A + m * KP + k0 + 2 * h);
      int pr = ((k0 >> 1) + h) * 128 + m * 2;
#pragma unroll
      for (int n = 0; n < 4; ++n) {
        v2f b = *(const v2f*)(sW0 + pr + n * 32);
        acc[n] = wmma4(a, b, acc[n]);
      }
    }
    // relu+bias, transpose hidden through the (reused) A region: H[16][64]
#pragma unroll
    for (int n = 0; n < 4; ++n) {
      float bb = sB0[n * 16 + m];
#pragma unroll
      for (int i = 0; i < 8; ++i)
        A[(i + 8 * h) * 64 + n * 16 + m] = fmaxf(acc[n][i] + bb, 0.0f);
    }
    v8f acc2[4] = {};
    for (int k0 = 0; k0 < 64; k0 += 4) {
      v2f a = *(const v2f*)(A + m * 64 + k0 + 2 * h);
      int pr = ((k0 >> 1) + h) * 128 + m * 2;
#pragma unroll
      for (int n = 0; n < 4; ++n) {
        v2f b = *(const v2f*)(sW1 + pr + n * 32);
        acc2[n] = wmma4(a, b, acc2[n]);
      }
    }
#pragma unroll
    for (int n = 0; n < 4; ++n) {
      int col = n * 16 + m;
      float bb = sB1[col];
#pragma unroll
      for (int i = 0; i < 8; ++i) {
        int g = base + i + 8 * h;
        if (g < N) out[(size_t)g * 64 + col] = fmaxf(acc2[n][i] + bb, 0.0f);
      }
    }
  }
}

// edge MLP: out = relu(relu([feats(5)|vc[vi](64)|cc[ci](64)]@W0+b0)@W1+b1)
// A columns remapped to [0..4 feats | 5..7 zero | 8..71 vc | 72..135 cc] (W0 rows permuted)
// 3 waves x 4 tiles x 16 edges per block; W1 pre-interleaved in ws (global b64 B-frags).
#define KE 136
__global__ __launch_bounds__(96) void edge_mlp_wmma_kernel(
    const float* __restrict__ vc, const float* __restrict__ cc,
    const float* __restrict__ lo, const float* __restrict__ hi,
    const float* __restrict__ dmm, const float* __restrict__ elp,
    const int* __restrict__ vi, const int* __restrict__ ci,
    const float* __restrict__ W0, const float* __restrict__ b0,
    const float* __restrict__ W1p, const float* __restrict__ b1,
    float* __restrict__ out, int E) {
  __shared__ alignas(16) float sW0[KE * 64];
  __shared__ float sB0[64], sB1[64];
  __shared__ alignas(16) float sA[3][16 * KE];
  int tid = threadIdx.x;
  for (int i = tid; i < KE * 64; i += 96) {
    int r = i >> 6, c = i & 63;
    float val = 0.0f;
    if (r < 5) val = W0[r * 64 + c];
    else if (r >= 8) val = W0[(r - 3) * 64 + c];
    sW0[(r >> 1) * 128 + c * 2 + (r & 1)] = val;  // pair-interleaved
  }
  if (tid < 64) { sB0[tid] = b0[tid]; sB1[tid] = b1[tid]; }
  int wave = tid >> 5, lane = tid & 31;
  int h = lane >> 4, m = lane & 15;
  float* A = sA[wave];
  __syncthreads();
  for (int t = 0; t < NTILES; ++t) {
    int base = ((blockIdx.x * 3 + wave) * NTILES + t) * 16;
    if (lane < 16) {
      int e = base + lane;
      float f0 = 0, f1 = 0, f2 = 0, f3 = 0, f4 = 0;
      if (e < E) { f0 = lo[e]; f1 = hi[e]; f2 = dmm[e]; f3 = elp[(size_t)e * 2]; f4 = elp[(size_t)e * 2 + 1]; }
      float* row = A + lane * KE;
      row[0] = f0; row[1] = f1; row[2] = f2; row[3] = f3; row[4] = f4;
      row[5] = 0.0f; row[6] = 0.0f; row[7] = 0.0f;
    }
#pragma unroll
    for (int tt = 0; tt < 8; ++tt) {  // vc rows, float4 (L2-resident gather)
      int id = tt * 32 + lane;
      int r = id >> 4, q = id & 15;
      int e = base + r;
      float4 vv = make_float4(0, 0, 0, 0);
      if (e < E) vv = *(const float4*)(vc + (size_t)vi[e] * 64 + q * 4);
      *(float4*)(A + r * KE + 8 + q * 4) = vv;
    }
#pragma unroll
    for (int tt = 0; tt < 8; ++tt) {  // cc rows
      int id = tt * 32 + lane;
      int r = id >> 4, q = id & 15;
      int e = base + r;
      float4 vv = make_float4(0, 0, 0, 0);
      if (e < E) vv = *(const float4*)(cc + (size_t)ci[e] * 64 + q * 4);
      *(float4*)(A + r * KE + 72 + q * 4) = vv;
    }
    v8f acc[4] = {};
    for (int k0 = 0; k0 < KE; k0 += 4) {
      v2f a = *(const v2f*)(A + m * KE + k0 + 2 * h);
      int pr = ((k0 >> 1) + h) * 128 + m * 2;
#pragma unroll
      for (int n = 0; n < 4; ++n) {
        v2f b = *(const v2f*)(sW0 + pr + n * 32);
        acc[n] = wmma4(a, b, acc[n]);
      }
    }
#pragma unroll
    for (int n = 0; n < 4; ++n) {
      float bb = sB0[n * 16 + m];
#pragma unroll
      for (int i = 0; i < 8; ++i)
        A[(i + 8 * h) * 64 + n * 16 + m] = fmaxf(acc[n][i] + bb, 0.0f);
    }
    v8f acc2[4] = {};
    for (int k0 = 0; k0 < 64; k0 += 4) {
      v2f a = *(const v2f*)(A + m * 64 + k0 + 2 * h);
      size_t pr = (size_t)((k0 >> 1) + h) * 128 + m * 2;
#pragma unroll
      for (int n = 0; n < 4; ++n) {
        v2f b = *(const v2f*)(W1p + pr + n * 32);  // pre-interleaved, one b64 per frag
        acc2[n] = wmma4(a, b, acc2[n]);
      }
    }
#pragma unroll
    for (int n = 0; n < 4; ++n) {
      int col = n * 16 + m;
      float bb = sB1[col];
#pragma unroll
      for (int i = 0; i < 8; ++i) {
        int g = base + i + 8 * h;
        if (g < E) out[(size_t)g * 64 + col] = fmaxf(acc2[n][i] + bb, 0.0f);
      }
    }
  }
}

// ---------------------------------------------------------------- host
static inline unsigned cdivu(long a, long b) { return (unsigned)((a + b - 1) / b); }

extern "C" void kernel_launch(void* const* d_in, const int* in_sizes, int n_in,
                              void* d_out, int out_size, void* d_ws, size_t ws_size,
                              hipStream_t stream) {
  const float* var_lp = (const float*)d_in[0];
  const float* con_lp = (const float*)d_in[1];
  const float* lo     = (const float*)d_in[2];
  const float* hi     = (const float*)d_in[3];
  const float* dmm    = (const float*)d_in[4];
  const float* elp    = (const float*)d_in[5];
  const int*   eidx   = (const int*)d_in[6];
  const int NV_ = in_sizes[0] / 2;
  const int NC_ = in_sizes[1] / 5;
  const int E_  = in_sizes[2];
  const int* vi = eidx;
  const int* ci = eidx + E_;

  const float* P[32];
  for (int i = 0; i < 32 && (7 + i) < n_in; ++i) P[i] = (const float*)d_in[7 + i];
  // params flattened in setup_inputs() dict insertion order (W then b per linear)
  const float *cq_W = P[0], *cq_b = P[1], *ck_W = P[2], *ck_b = P[3], *cv_W = P[4], *cv_b = P[5],
              *ce_W = P[6], *ce_b = P[7], *cs_W = P[8], *cs_b = P[9];
  const float *vq_W = P[10], *vq_b = P[11], *vk_W = P[12], *vk_b = P[13], *vv_W = P[14],
              *vv_b = P[15], *ve_W = P[16], *ve_b = P[17], *vs_W = P[18], *vs_b = P[19];
  const float *uv_W0 = P[20], *uv_b0 = P[21], *uv_W1 = P[22], *uv_b1 = P[23];
  const float *uc_W0 = P[24], *uc_b0 = P[25], *uc_W1 = P[26], *uc_b1 = P[27];
  const float *ue_W0 = P[28], *ue_b0 = P[29], *ue_W1 = P[30], *ue_b1 = P[31];

  float* Wb = (float*)d_ws;
  float* logits = Wb;                        // [E]
  float* w1p    = Wb + (size_t)E_;           // [64*64] interleaved edge W1
  size_t Pof = (size_t)E_ + 64 * 64;
  // phase A scratch (reused by later phases)
  float* qc     = Wb + Pof;
  float* numerA = qc + (size_t)NC_ * 64;
  float* denomA = numerA + (size_t)NC_ * 64;
  float* degA   = denomA + NC_;
  float* mA     = degA + NC_;
  // phase B scratch (overlays phase A)
  float* k2     = Wb + Pof;
  float* v2     = k2 + (size_t)NC_ * 64;
  float* numerB = v2 + (size_t)NC_ * 64;
  float* denomB = numerB + (size_t)NV_ * 64;
  float* degB   = denomB + NV_;
  float* mB     = degB + NV_;
  // phase C scratch (overlays phase B)
  float* vc = Wb + Pof;
  float* cc = vc + (size_t)NV_ * 64;

  float* out_var  = (float*)d_out;                  // [NV,64]
  float* out_con  = out_var + (size_t)NV_ * 64;     // [NC,64]
  float* out_edge = out_con + (size_t)NC_ * 64;     // [E,64]

  interleave_w_kernel<<<16, 256, 0, stream>>>(ue_W1, w1p);

  // ---- Phase A: con_conv ----
  fill_kernel<<<cdivu((long)NC_ * 64 + 2 * NC_, 256), 256, 0, stream>>>(numerA, 0.0f, (long)NC_ * 64 + 2 * NC_);
  fill_kernel<<<cdivu(NC_, 256), 256, 0, stream>>>(mA, -INFINITY, NC_);
  node_linear_kernel<<<cdivu((long)NC_ * 64, 256), 256, 0, stream>>>(con_lp, 5, cq_W, cq_b, qc, (long)NC_ * 64);
  con_pass1_kernel<<<cdivu(E_, 256), 256, 0, stream>>>(var_lp, lo, hi, dmm, elp, vi, ci, qc,
                                                       ck_W, ck_b, ce_W, ce_b, logits, mA, degA, E_);
  con_pass2_kernel<<<cdivu(E_, 256), 256, 0, stream>>>(var_lp, lo, hi, dmm, elp, vi, ci,
                                                       cv_W, cv_b, ce_W, ce_b, logits, mA, numerA, denomA, E_);
  finalize_kernel<<<cdivu((long)NC_ * 64, 256), 256, 0, stream>>>(numerA, denomA, degA, con_lp, 5,
                                                                  cs_W, cs_b, out_con, (long)NC_ * 64);
  // ---- Phase B: var_conv ----
  dual_linear_wmma_kernel<<<cdivu(NC_, 256), 128, 0, stream>>>(out_con, con_lp, 5, NC_,
                                                               vk_W, vk_b, vv_W, vv_b, k2, v2);
  fill_kernel<<<cdivu((long)NV_ * 64 + 2 * NV_, 256), 256, 0, stream>>>(numerB, 0.0f, (long)NV_ * 64 + 2 * NV_);
  fill_kernel<<<cdivu(NV_, 256), 256, 0, stream>>>(mB, -INFINITY, NV_);
  var_pass1_kernel<<<cdivu(E_, 256), 256, 0, stream>>>(var_lp, lo, hi, dmm, elp, vi, ci, k2,
                                                       vq_W, vq_b, ve_W, ve_b, logits, mB, degB, E_);
  var_pass2_kernel<<<cdivu(E_, 256), 256, 0, stream>>>(lo, hi, dmm, elp, vi, ci, v2,
                                                       ve_W, ve_b, logits, mB, numerB, denomB, E_);
  finalize_kernel<<<cdivu((long)NV_ * 64, 256), 256, 0, stream>>>(numerB, denomB, degB, var_lp, 2,
                                                                  vs_W, vs_b, out_var, (long)NV_ * 64);
  // ---- Phase C: node MLPs (WMMA) ----
  mlp2_wmma_kernel<<<cdivu(NV_, 256), 128, 0, stream>>>(out_var, var_lp, 2, NV_, uv_W0, uv_b0, uv_W1, uv_b1, vc);
  mlp2_wmma_kernel<<<cdivu(NC_, 256), 128, 0, stream>>>(out_con, con_lp, 5, NC_, uc_W0, uc_b0, uc_W1, uc_b1, cc);
  // ---- Phase D: edge MLP (WMMA, dominant GEMM) ----
  edge_mlp_wmma_kernel<<<cdivu(E_, 192), 96, 0, stream>>>(vc, cc, lo, hi, dmm, elp, vi, ci,
                                                          ue_W0, ue_b0, w1p, ue_b1, out_edge, E_);
}